// GVIN_47811575939382
// MI455X (gfx1250) — compile-verified
//
#include <hip/hip_runtime.h>
#include <math.h>

// Problem constants (match reference)
static constexpr int   NN    = 4096;
static constexpr int   DIN   = 16;
static constexpr int   CCH   = 16;
static constexpr int   CHQ   = 8;
static constexpr int   KSTEP = 10;
static constexpr float GAMMA = 0.99f;
static constexpr float EPSF  = 1.1920928955078125e-07f; // np.finfo(f32).eps

typedef float v2f __attribute__((ext_vector_type(2)));
typedef float v8f __attribute__((ext_vector_type(8)));

// ---------------------------------------------------------------------------
// Zero the accumulators / degree / value arrays (workspace is poisoned 0xAA).
__global__ void k_init(float* __restrict__ deg, float* __restrict__ yp,
                       float* __restrict__ yw, float* __restrict__ v) {
  int i = blockIdx.x * blockDim.x + threadIdx.x;
  if (i < NN) { deg[i] = 0.f; yp[i] = 0.f; yw[i] = 0.f; v[i] = 0.f; }
}

// ---------------------------------------------------------------------------
// deg[j] = sum_i adj[i,j]  (the +1 from A+I is folded in later).
__global__ void __launch_bounds__(256)
k_deg(const float* __restrict__ adj, float* __restrict__ deg) {
  int j  = blockIdx.x * 256 + threadIdx.x;
  int i0 = blockIdx.y * 256;
  const float* p = adj + (size_t)i0 * NN + j;
  float sum = 0.f;
#pragma unroll 8
  for (int i = 0; i < 256; ++i) sum += p[(size_t)i * NN];
  atomicAdd(&deg[j], sum);
}

// ---------------------------------------------------------------------------
// Per-node features: r = xc@Wr+br ; s = (xc@We+be)@w_emb ; dinv = sqrt(1/(deg+1+eps))
// Also seeds p/w for iteration 0 (v=0 -> u=r).
__global__ void __launch_bounds__(256)
k_node(const float* __restrict__ x, const float* __restrict__ comms,
       const float* __restrict__ Wr, const float* __restrict__ br,
       const float* __restrict__ We, const float* __restrict__ be,
       const float* __restrict__ w_emb, const float* __restrict__ deg,
       float* __restrict__ r, float* __restrict__ s, float* __restrict__ dinv,
       float* __restrict__ gp, float* __restrict__ gw) {
  int i = blockIdx.x * 256 + threadIdx.x;
  if (i >= NN) return;
  float xc[DIN + CCH];
#pragma unroll
  for (int t = 0; t < DIN; ++t) xc[t] = x[i * DIN + t];
#pragma unroll
  for (int t = 0; t < CCH; ++t) xc[DIN + t] = comms[i * CCH + t];

  float ri = br[0];
#pragma unroll
  for (int t = 0; t < DIN + CCH; ++t) ri += xc[t] * Wr[t];

  float si = 0.f;
#pragma unroll
  for (int ch = 0; ch < CHQ; ++ch) {
    float e = be[ch];
#pragma unroll
    for (int t = 0; t < DIN + CCH; ++t) e += xc[t] * We[t * CHQ + ch];
    si += e * w_emb[ch];
  }
  float di = sqrtf(1.0f / (deg[i] + 1.0f + EPSF));
  r[i] = ri; s[i] = si; dinv[i] = di;
  gp[i] = di * si * ri;   // p = dinv * s * u,  u = r + gamma*0
  gw[i] = di * ri;        // w = dinv * u
}

// ---------------------------------------------------------------------------
// Mat-vec core: Y[c,i] += sum_k Z[c,k] * adj[k,i]  via V_WMMA_F32_16X16X4_F32.
// Z row 0 = p, row 1 = w, rows 2..15 = 0; adj symmetric so this equals adj@z.
// Block = 256 thr = 8 waves; each wave: 16 output columns, K-chunk of 256.
// A operand comes from LDS (p chunk | w chunk | zero pad) so every lane does an
// unconditional ds_load_b64 — no exec divergence. Inner loop is batched 8-deep
// (16 global b32 + 8 ds b64 in flight) before the 8 chained WMMAs.
__global__ void __launch_bounds__(256)
k_mv(const float* __restrict__ adj, const float* __restrict__ gp,
     const float* __restrict__ gw, float* __restrict__ yp,
     float* __restrict__ yw) {
  __shared__ float sz[3 * 256];        // [0,256): p  [256,512): w  [512,768): 0
  const int tid  = threadIdx.x;
  const int lane = tid & 31;
  const int wave = tid >> 5;
  const int c    = lane & 15;          // A: channel (row of Z) ; B: column offset
  const int kk   = (lane >> 4) << 1;   // 0 for lanes 0-15, 2 for lanes 16-31
  const int i0   = blockIdx.x * 128 + wave * 16;
  const int kb   = blockIdx.y * 256;

  // Cooperative staging of this block's K-chunk of p/w plus zero pad.
  sz[tid]       = gp[kb + tid];
  sz[256 + tid] = gw[kb + tid];
  sz[512 + tid] = 0.f;
  __syncthreads();

  const float* ap = sz + ((c == 0) ? 0 : (c == 1) ? 256 : 512);
  const float* bp = adj + (size_t)(kb + kk) * NN + (i0 + c);

  v8f acc = {};
#pragma unroll 1
  for (int t0 = 0; t0 < 64; t0 += 8) {
    v2f a[8], b[8];
#pragma unroll
    for (int u = 0; u < 8; ++u) {
      const int t = t0 + u;
      const int k = t * 4 + kk;
      b[u].x = bp[(size_t)(t * 4) * NN];        // adj row k   , cols i0..i0+15
      b[u].y = bp[(size_t)(t * 4 + 1) * NN];    // adj row k+1
      a[u].x = ap[k];                           // ds_load_b64 (8B aligned)
      a[u].y = ap[k + 1];
    }
#pragma unroll
    for (int u = 0; u < 8; ++u)
      acc = __builtin_amdgcn_wmma_f32_16x16x4_f32(
          false, a[u], false, b[u], (short)0, acc, false, false);
  }
  // D layout: V0 lanes0-15 = Y[0, i0+lane] (=yp), V1 lanes0-15 = Y[1, ...] (=yw)
  if (lane < 16) {
    atomicAdd(&yp[i0 + lane], acc[0]);
    atomicAdd(&yw[i0 + lane], acc[1]);
  }
}

// ---------------------------------------------------------------------------
// Per-node combine: k3v -> action decode -> max -> new v; prep next p/w,
// reset accumulators; on the last step apply the -inf mask and write output.
__global__ void __launch_bounds__(256)
k_combine(const float* __restrict__ r, const float* __restrict__ s,
          const float* __restrict__ dinv, float* __restrict__ v,
          float* __restrict__ yp, float* __restrict__ yw,
          float* __restrict__ gp, float* __restrict__ gw,
          const float* __restrict__ Wa, const float* __restrict__ ba,
          const float* __restrict__ b_emb, const float* __restrict__ mask,
          float* __restrict__ out, int last) {
  int i = blockIdx.x * 256 + threadIdx.x;
  if (i >= NN) return;
  float ri = r[i], si = s[i], di = dinv[i], vi = v[i];
  float u  = ri + GAMMA * vi;           // same u that built this iteration's p/w
  float pi = di * si * u;
  float wi = di * u;
  float k3v = di * ((yp[i] + pi) + (b_emb[0] - si) * (yw[i] + wi));
  float vmax = -INFINITY;
#pragma unroll
  for (int ch = 0; ch < CHQ; ++ch) vmax = fmaxf(vmax, k3v * Wa[ch] + ba[ch]);
  v[i] = vmax;
  yp[i] = 0.f; yw[i] = 0.f;             // reset accumulators for next iteration
  float un = ri + GAMMA * vmax;         // next iteration's u
  gp[i] = di * si * un;
  gw[i] = di * un;
  if (last) out[i] = vmax + ((mask[i] == 0.0f) ? -INFINITY : 0.0f);
}

// ---------------------------------------------------------------------------
extern "C" void kernel_launch(void* const* d_in, const int* in_sizes, int n_in,
                              void* d_out, int out_size, void* d_ws,
                              size_t ws_size, hipStream_t stream) {
  const float* x     = (const float*)d_in[0];
  const float* comms = (const float*)d_in[1];
  const float* adj   = (const float*)d_in[2];
  const float* mask  = (const float*)d_in[3];
  const float* Wr    = (const float*)d_in[4];
  const float* br    = (const float*)d_in[5];
  const float* We    = (const float*)d_in[6];
  const float* be    = (const float*)d_in[7];
  const float* wemb  = (const float*)d_in[8];
  const float* bemb  = (const float*)d_in[9];
  const float* Wa    = (const float*)d_in[10];
  const float* ba    = (const float*)d_in[11];
  float* out = (float*)d_out;

  float* ws   = (float*)d_ws;
  float* deg  = ws + 0 * NN;
  float* dinv = ws + 1 * NN;
  float* r    = ws + 2 * NN;
  float* s    = ws + 3 * NN;
  float* v    = ws + 4 * NN;
  float* yp   = ws + 5 * NN;
  float* yw   = ws + 6 * NN;
  float* gp   = ws + 7 * NN;
  float* gw   = ws + 8 * NN;

  k_init<<<NN / 256, 256, 0, stream>>>(deg, yp, yw, v);
  k_deg<<<dim3(NN / 256, NN / 256), 256, 0, stream>>>(adj, deg);
  k_node<<<NN / 256, 256, 0, stream>>>(x, comms, Wr, br, We, be, wemb, deg,
                                       r, s, dinv, gp, gw);
  for (int t = 0; t < KSTEP; ++t) {
    k_mv<<<dim3(NN / 128, NN / 256), 256, 0, stream>>>(adj, gp, gw, yp, yw);
    k_combine<<<NN / 256, 256, 0, stream>>>(r, s, dinv, v, yp, yw, gp, gw,
                                            Wa, ba, bemb, mask, out,
                                            (t == KSTEP - 1) ? 1 : 0);
  }
}